// GraphVAE_20555713479114
// MI455X (gfx1250) — compile-verified
//
#include <hip/hip_runtime.h>
#include <hip/hip_bf16.h>
#include <math.h>

#define NN 96
#define NSQ (NN*NN)          // 9216
#define LATENT 256
#define OUT_DIM 4656         // 96*97/2
#define MPM_ITERS 50

typedef float v2f __attribute__((ext_vector_type(2)));
typedef float v8f __attribute__((ext_vector_type(8)));

// ---------------------------------------------------------------------------
// Encoder GEMVs: mu = h@W_mu+b_mu ; logstd = h@W_ls+b_ls  (block 0 / block 1)
// Thread c owns output column c -> W reads fully coalesced across the block.
// ---------------------------------------------------------------------------
__global__ __launch_bounds__(256) void gemv_enc(
    const float* __restrict__ h,
    const float* __restrict__ Wmu, const float* __restrict__ bmu,
    const float* __restrict__ Wls, const float* __restrict__ bls,
    float* __restrict__ mu, float* __restrict__ ls)
{
    const int c = threadIdx.x;
    const float* __restrict__ W = (blockIdx.x == 0) ? Wmu : Wls;
    const float* __restrict__ b = (blockIdx.x == 0) ? bmu : bls;
    float* __restrict__ o       = (blockIdx.x == 0) ? mu  : ls;
    float acc = b[c];
    for (int k = 0; k < NSQ; ++k)
        acc += h[k] * W[k * LATENT + c];
    o[c] = acc;
}

// KL = -0.5 * sum(1 + logstd - mu^2 - exp(logstd)) / (N*N)
__global__ __launch_bounds__(256) void kl_kernel(
    const float* __restrict__ mu, const float* __restrict__ ls,
    float* __restrict__ kl)
{
    __shared__ float red[256];
    const int t = threadIdx.x;
    float m = mu[t], l = ls[t];
    red[t] = 1.0f + l - m * m - expf(l);
    __syncthreads();
    for (int s = 128; s > 0; s >>= 1) {
        if (t < s) red[t] += red[t + s];
        __syncthreads();
    }
    if (t == 0) kl[0] = -0.5f * red[0] / (float)NSQ;
}

// y = relu(concat(h, mu) @ W_d1 + b_d1)
__global__ __launch_bounds__(256) void gemv_d1(
    const float* __restrict__ h, const float* __restrict__ mu,
    const float* __restrict__ W, const float* __restrict__ b,
    float* __restrict__ y)
{
    const int c = threadIdx.x;
    float acc = b[c];
    for (int k = 0; k < NSQ; ++k)
        acc += h[k] * W[k * LATENT + c];
    for (int k = 0; k < LATENT; ++k)
        acc += mu[k] * W[(NSQ + k) * LATENT + c];
    y[c] = fmaxf(acc, 0.0f);
}

// out = sigmoid(y @ W_d2 + b_d2)   (4656 outputs)
__global__ __launch_bounds__(256) void gemv_d2(
    const float* __restrict__ y, const float* __restrict__ W,
    const float* __restrict__ b, float* __restrict__ out)
{
    __shared__ float ys[LATENT];
    ys[threadIdx.x] = y[threadIdx.x];
    __syncthreads();
    const int o = blockIdx.x * 256 + threadIdx.x;
    if (o < OUT_DIM) {
        float acc = b[o];
        for (int k = 0; k < LATENT; ++k)
            acc += ys[k] * W[k * OUT_DIM + o];
        out[o] = 1.0f / (1.0f + expf(-acc));
    }
}

// rec = symmetrize(upper-triangular scatter of out)
__global__ void build_rec(const float* __restrict__ out, float* __restrict__ rec)
{
    const int t = blockIdx.x * 256 + threadIdx.x;
    if (t >= NSQ) return;
    const int r = t / NN, c = t - r * NN;
    if (r <= c) {
        const int o = r * NN - (r * (r - 1)) / 2 + (c - r);
        const float v = out[o];
        rec[r * NN + c] = v;
        rec[c * NN + r] = v;
    }
}

// row sums + diagonals of adj and rec
__global__ void rowsums(const float* __restrict__ adj, const float* __restrict__ rec,
                        float* __restrict__ nf, float* __restrict__ nfr,
                        float* __restrict__ dd, float* __restrict__ ddr)
{
    const int i = threadIdx.x;
    if (i < NN) {
        float s1 = 0.0f, s2 = 0.0f;
        for (int j = 0; j < NN; ++j) { s1 += adj[i * NN + j]; s2 += rec[i * NN + j]; }
        nf[i] = s1; nfr[i] = s2;
        dd[i] = adj[i * (NN + 1)];
        ddr[i] = rec[i * (NN + 1)];
    }
}

// A = adj*d_i*d_j*noteye ; B = rec*dr_i*dr_j*noteye ; D = d_i*dr_j/(|nf_i-nfr_j|+1)
__global__ void build_abd(const float* __restrict__ adj, const float* __restrict__ rec,
                          const float* __restrict__ nf, const float* __restrict__ nfr,
                          const float* __restrict__ dd, const float* __restrict__ ddr,
                          float* __restrict__ A, float* __restrict__ B, float* __restrict__ D)
{
    const int t = blockIdx.x * 256 + threadIdx.x;
    if (t >= NSQ) return;
    const int i = t / NN, j = t - i * NN;
    const float ne = (i != j) ? 1.0f : 0.0f;
    A[t] = adj[t] * dd[i] * dd[j] * ne;
    B[t] = rec[t] * ddr[i] * ddr[j] * ne;
    D[t] = dd[i] * ddr[j] / (fabsf(nf[i] - nfr[j]) + 1.0f);
}

// ---------------------------------------------------------------------------
// MPM: 50 iterations of xn = x*D + A @ (max-pool_b x[j,b]*B[a,b]), normalize.
// Single persistent 512-thread workgroup; x, B, P live in LDS (~110 KB).
// A@P uses V_WMMA_F32_16X16X4_F32 (exact fp32) over 36 tiles / 16 waves.
// ---------------------------------------------------------------------------
__global__ __launch_bounds__(512) void mpm_kernel(
    const float* __restrict__ Ag, const float* __restrict__ Bg,
    const float* __restrict__ Dg, float* __restrict__ xout)
{
    extern __shared__ float smem[];
    float* x_s = smem;              // 9216
    float* B_s = smem + NSQ;        // 9216
    float* P_s = smem + 2 * NSQ;    // 9216
    __shared__ float red[512];
    __shared__ float scale_s;

    const int tid  = threadIdx.x;
    const int lane = tid & 31;
    const int wave = tid >> 5;      // 0..15

    for (int i = tid; i < NSQ; i += 512) {
        x_s[i] = 1.0f / (float)NN;
        B_s[i] = Bg[i];
    }
    __syncthreads();

    for (int it = 0; it < MPM_ITERS; ++it) {
        // Phase 1: P[j,a] = max_b x[j,b]*B[b,a]  (B symmetric; [b*N+a] indexing
        // keeps consecutive lanes on distinct LDS banks)
        for (int o = tid; o < NSQ; o += 512) {
            const int j = o / NN, a = o - j * NN;
            const float* __restrict__ xr = &x_s[j * NN];
            float m = -3.0e38f;
            for (int b = 0; b < NN; ++b)
                m = fmaxf(m, xr[b] * B_s[b * NN + a]);
            P_s[o] = m;
        }
        __syncthreads();

        // Phase 2: neigh = A @ P via WMMA; immediately fuse xn = acc + x*D
        // into x_s (each index written by exactly its owning lane; x_s is not
        // otherwise read in this phase, so no extra barrier is needed).
        const int m16  = lane & 15;
        const int kh   = lane >> 4;      // 0 or 1
        for (int t = wave; t < 36; t += 16) {   // wave-uniform -> EXEC all 1s
            const int ti = t / 6, tj = t - ti * 6;
            v8f c = {};
            for (int kb = 0; kb < NN; kb += 4) {
                v2f af, bf;
                const int arow = ti * 16 + m16;
                af.x = Ag[arow * NN + kb + 2 * kh];
                af.y = Ag[arow * NN + kb + 2 * kh + 1];
                const int bcol = tj * 16 + m16;
                bf.x = P_s[(kb + 2 * kh)     * NN + bcol];
                bf.y = P_s[(kb + 2 * kh + 1) * NN + bcol];
                c = __builtin_amdgcn_wmma_f32_16x16x4_f32(
                        false, af, false, bf, (short)0, c, false, false);
            }
            const int col = tj * 16 + m16;
#pragma unroll
            for (int r = 0; r < 8; ++r) {
                const int row = ti * 16 + r + 8 * kh;
                const int idx = row * NN + col;
                x_s[idx] = c[r] + x_s[idx] * Dg[idx];
            }
        }
        __syncthreads();

        // Phase 3: Frobenius norm, x = xn / ||xn||
        float ss = 0.0f;
        for (int o = tid; o < NSQ; o += 512) { const float v = x_s[o]; ss += v * v; }
        red[tid] = ss;
        __syncthreads();
        for (int s = 256; s > 0; s >>= 1) {
            if (tid < s) red[tid] += red[tid + s];
            __syncthreads();
        }
        if (tid == 0) scale_s = rsqrtf(red[0]);
        __syncthreads();
        for (int o = tid; o < NSQ; o += 512) x_s[o] *= scale_s;
        __syncthreads();
    }
    for (int o = tid; o < NSQ; o += 512) xout[o] = x_s[o];
}

// ---------------------------------------------------------------------------
// Hungarian (min-cost assignment) on cost = -x, double precision like the
// reference. One 128-thread block; column relax + argmin lane-parallel.
// Produces ind with ind[col_of_row[r]] = r.
// ---------------------------------------------------------------------------
__global__ __launch_bounds__(128) void hungarian_kernel(
    const float* __restrict__ xg, int* __restrict__ ind)
{
    __shared__ float cost[NSQ];
    __shared__ double u[NN + 1], v[NN + 1], minv[NN + 1];
    __shared__ int p[NN + 1], way[NN + 1], used[NN + 1];
    __shared__ double rv[128];
    __shared__ int ri[128];
    __shared__ int j0s;
    __shared__ int colrow[NN];
    const int tid = threadIdx.x;

    for (int t = tid; t < NSQ; t += 128) cost[t] = -xg[t];
    if (tid <= NN) { u[tid] = 0.0; v[tid] = 0.0; p[tid] = 0; way[tid] = 0; }
    __syncthreads();

    for (int i = 1; i <= NN; ++i) {
        if (tid == 0) { p[0] = i; j0s = 0; }
        if (tid <= NN) { minv[tid] = 1.0e18; used[tid] = 0; }
        __syncthreads();
        while (true) {
            const int j0 = j0s;
            if (tid == 0) used[j0] = 1;
            __syncthreads();
            const int i0 = p[j0];
            if (tid < NN) {
                const int j = tid + 1;
                if (!used[j]) {
                    const double cur = (double)cost[(i0 - 1) * NN + (j - 1)] - u[i0] - v[j];
                    if (cur < minv[j]) { minv[j] = cur; way[j] = j0; }
                }
            }
            __syncthreads();
            // argmin over free columns; tie -> smallest j (np.argmin semantics)
            double bv = 1.0e18; int bi = 9999;
            if (tid < NN) { const int j = tid + 1; if (!used[j]) { bv = minv[j]; bi = j; } }
            rv[tid] = bv; ri[tid] = bi;
            __syncthreads();
            for (int s = 64; s > 0; s >>= 1) {
                if (tid < s) {
                    if (rv[tid + s] < rv[tid] ||
                        (rv[tid + s] == rv[tid] && ri[tid + s] < ri[tid])) {
                        rv[tid] = rv[tid + s]; ri[tid] = ri[tid + s];
                    }
                }
                __syncthreads();
            }
            const int j1 = ri[0];
            const double delta = rv[0];
            __syncthreads();
            if (tid <= NN) {
                if (used[tid]) { u[p[tid]] += delta; v[tid] -= delta; }
                else if (tid >= 1) minv[tid] -= delta;
            }
            if (tid == 0) j0s = j1;
            __syncthreads();
            if (p[j1] == 0) break;   // uniform: shared p, uniform j1
        }
        if (tid == 0) {              // augment along "way" chain
            int j0a = j0s;
            while (j0a) { const int j1a = way[j0a]; p[j0a] = p[j1a]; j0a = j1a; }
        }
        __syncthreads();
    }
    if (tid < NN) colrow[p[tid + 1] - 1] = tid;   // col_of_row
    __syncthreads();
    if (tid < NN) ind[colrow[tid]] = tid;          // ind[col_ind] = arange
}

// BCE(adj permuted, out) mean + KL -> scalar
__global__ __launch_bounds__(256) void final_loss(
    const float* __restrict__ adj, const float* __restrict__ out,
    const int* __restrict__ ind, const float* __restrict__ kl,
    float* __restrict__ d_out)
{
    __shared__ float red[256];
    const int tid = threadIdx.x;
    float acc = 0.0f;
    for (int t = tid; t < NSQ; t += 256) {
        const int r = t / NN, c = t - r * NN;
        if (r <= c) {
            const int o = r * NN - (r * (r - 1)) / 2 + (c - r);
            const float a = adj[ind[r] * NN + ind[c]];
            const float tt = out[o];
            acc += fmaxf(a, 0.0f) - a * tt + log1pf(expf(-fabsf(a)));
        }
    }
    red[tid] = acc;
    __syncthreads();
    for (int s = 128; s > 0; s >>= 1) {
        if (tid < s) red[tid] += red[tid + s];
        __syncthreads();
    }
    if (tid == 0) d_out[0] = red[0] / (float)OUT_DIM + kl[0];
}

// ---------------------------------------------------------------------------
extern "C" void kernel_launch(void* const* d_in, const int* in_sizes, int n_in,
                              void* d_out, int out_size, void* d_ws, size_t ws_size,
                              hipStream_t stream)
{
    const float* h    = (const float*)d_in[0];   // input_features (9216)
    const float* adj  = (const float*)d_in[1];   // adj (1,96,96)
    const float* W_mu = (const float*)d_in[2];
    const float* b_mu = (const float*)d_in[3];
    const float* W_ls = (const float*)d_in[4];
    const float* b_ls = (const float*)d_in[5];
    const float* W_d1 = (const float*)d_in[6];
    const float* b_d1 = (const float*)d_in[7];
    const float* W_d2 = (const float*)d_in[8];
    const float* b_d2 = (const float*)d_in[9];

    float* ws = (float*)d_ws;
    float* mu   = ws;                 // 256
    float* ls   = ws + 256;           // 256
    float* y    = ws + 512;           // 256
    float* outv = ws + 768;           // 4656
    float* rec  = ws + 5424;          // 9216
    float* Am   = ws + 14640;         // 9216
    float* Bm   = ws + 23856;         // 9216
    float* Dm   = ws + 33072;         // 9216
    float* xm   = ws + 42288;         // 9216
    float* nf   = ws + 51504;         // 96
    float* nfr  = ws + 51600;         // 96
    float* dd   = ws + 51696;         // 96
    float* ddr  = ws + 51792;         // 96
    float* kl   = ws + 51888;         // 1
    int*   ind  = (int*)(ws + 51904); // 96

    gemv_enc<<<2, 256, 0, stream>>>(h, W_mu, b_mu, W_ls, b_ls, mu, ls);
    kl_kernel<<<1, 256, 0, stream>>>(mu, ls, kl);
    gemv_d1<<<1, 256, 0, stream>>>(h, mu, W_d1, b_d1, y);
    gemv_d2<<<(OUT_DIM + 255) / 256, 256, 0, stream>>>(y, W_d2, b_d2, outv);
    build_rec<<<(NSQ + 255) / 256, 256, 0, stream>>>(outv, rec);
    rowsums<<<1, 128, 0, stream>>>(adj, rec, nf, nfr, dd, ddr);
    build_abd<<<(NSQ + 255) / 256, 256, 0, stream>>>(adj, rec, nf, nfr, dd, ddr,
                                                     Am, Bm, Dm);
    mpm_kernel<<<1, 512, 3 * NSQ * sizeof(float), stream>>>(Am, Bm, Dm, xm);
    hungarian_kernel<<<1, 128, 0, stream>>>(xm, ind);
    final_loss<<<1, 256, 0, stream>>>(adj, outv, ind, kl, (float*)d_out);
}